// FocalLoss_68831145886018
// MI455X (gfx1250) — compile-verified
//
#include <hip/hip_runtime.h>

typedef __attribute__((ext_vector_type(2))) float v2f;
typedef __attribute__((ext_vector_type(4))) float v4f;
typedef __attribute__((ext_vector_type(8))) float v8f;
typedef __attribute__((ext_vector_type(4))) int   v4i;

#define F_ALPHA 0.7f
#define F_EPS   1e-7f
#define NBLOCKS 2048
#define NTHREADS 256

// -ln(2): converts v_log_f32 (log2) to -natural-log in one multiply
#define NEG_LN2 (-0.69314718055994531f)

__device__ __forceinline__ float fast_neg_log(float x) {
#if __has_builtin(__builtin_amdgcn_logf)
    return __builtin_amdgcn_logf(x) * NEG_LN2;   // v_log_f32 + v_mul
#else
    return -__logf(x);
#endif
}

__device__ __forceinline__ float focal_elem(float p, int y) {
    const bool pos = (y == 1);
    float pt = pos ? p : 1.0f - p;
    float l = fast_neg_log(pt);                  // +inf iff pt == 0
    if (__builtin_isinf(l)) l = F_EPS;           // reference: inf -> 1e-7
    float om = 1.0f - pt;
    float f = om * om * l;                       // GAMMA == 2
    return pos ? f * F_ALPHA : f;
}

// Reduce the 32 per-lane partial pairs (acc0, acc1) of a wave to the wave sum,
// broadcast to all lanes, using V_WMMA_F32_16X16X4_F32 with B = ones.
// A layout (16x4 f32): lanes 0-15 hold {K0,K1} for row M=lane, lanes 16-31 hold
// {K2,K3} for row M=lane-16.  With B=1: D[m][n] = acc0_m+acc1_m+acc0_{m+16}+acc1_{m+16}.
// D layout: vgpr r, lanes 0-15 -> M=r, lanes 16-31 -> M=8+r.  Summing the 8 D
// vgprs gives rows 0-7 (lanes 0-15) / rows 8-15 (lanes 16-31); xor-16 finishes.
__device__ __forceinline__ float wave_reduce(float acc0, float acc1) {
#if __has_builtin(__builtin_amdgcn_wmma_f32_16x16x4_f32)
    v2f a; a[0] = acc0; a[1] = acc1;
    v2f b; b[0] = 1.0f; b[1] = 1.0f;
    v8f c = {};
    v8f d = __builtin_amdgcn_wmma_f32_16x16x4_f32(
        /*neg_a=*/false, a, /*neg_b=*/false, b,
        /*c_mod=*/(short)0, c, /*reuse_a=*/false, /*reuse_b=*/false);
    float s = ((d[0] + d[1]) + (d[2] + d[3])) + ((d[4] + d[5]) + (d[6] + d[7]));
    s += __shfl_xor(s, 16, 32);
    return s;
#else
    float s = acc0 + acc1;
    for (int off = 16; off > 0; off >>= 1) s += __shfl_xor(s, off, 32);
    return s;
#endif
}

__global__ __launch_bounds__(NTHREADS)
void focal_partial_kernel(const float* __restrict__ p, const int* __restrict__ y,
                          float* __restrict__ partial, int n) {
    const int nvec   = n >> 2;
    const int gid    = blockIdx.x * NTHREADS + threadIdx.x;
    const int stride = gridDim.x * NTHREADS;
    const v4f* __restrict__ p4 = (const v4f*)p;
    const v4i* __restrict__ y4 = (const v4i*)y;

    float acc0 = 0.0f, acc1 = 0.0f;
    #pragma unroll 2
    for (int i = gid; i < nvec; i += stride) {
        v4f pv = __builtin_nontemporal_load(&p4[i]);   // global_load_b128, TH=NT
        v4i yv = __builtin_nontemporal_load(&y4[i]);
        acc0 += focal_elem(pv.x, yv.x);
        acc0 += focal_elem(pv.y, yv.y);
        acc1 += focal_elem(pv.z, yv.z);
        acc1 += focal_elem(pv.w, yv.w);
    }
    if (gid == 0) {                                    // tail (n % 4)
        for (int i = nvec << 2; i < n; ++i) acc0 += focal_elem(p[i], y[i]);
    }

    // all 256 threads reach here uniformly -> EXEC all-1s for the WMMA
    float wsum = wave_reduce(acc0, acc1);

    __shared__ float lds[NTHREADS / 32];
    const int lane = threadIdx.x & 31;
    const int wave = threadIdx.x >> 5;
    if (lane == 0) lds[wave] = wsum;
    __syncthreads();
    if (threadIdx.x == 0) {
        float s = lds[0];
        #pragma unroll
        for (int w = 1; w < NTHREADS / 32; ++w) s += lds[w];
        partial[blockIdx.x] = s;
    }
}

__global__ __launch_bounds__(NTHREADS)
void focal_final_kernel(const float* __restrict__ partial, float* __restrict__ out,
                        int nblocks, double inv_n) {
    float s = 0.0f;
    for (int i = threadIdx.x; i < nblocks; i += NTHREADS) s += partial[i];
    for (int off = 16; off > 0; off >>= 1) s += __shfl_xor(s, off, 32);

    __shared__ float lds[NTHREADS / 32];
    const int lane = threadIdx.x & 31;
    const int wave = threadIdx.x >> 5;
    if (lane == 0) lds[wave] = s;
    __syncthreads();
    if (threadIdx.x == 0) {
        float t = lds[0];
        #pragma unroll
        for (int w = 1; w < NTHREADS / 32; ++w) t += lds[w];
        out[0] = (float)((double)t * inv_n);           // mean
    }
}

extern "C" void kernel_launch(void* const* d_in, const int* in_sizes, int n_in,
                              void* d_out, int out_size, void* d_ws, size_t ws_size,
                              hipStream_t stream) {
    const float* p = (const float*)d_in[0];   // sigmoid probs, float32
    const int*   y = (const int*)d_in[1];     // binary targets, int32
    float* out     = (float*)d_out;
    float* partial = (float*)d_ws;            // NBLOCKS * 4 bytes of scratch
    const int n = in_sizes[0];

    int nblocks = NBLOCKS;
    if (ws_size < (size_t)nblocks * sizeof(float)) {
        nblocks = (int)(ws_size / sizeof(float));
        if (nblocks < 1) nblocks = 1;
    }

    focal_partial_kernel<<<nblocks, NTHREADS, 0, stream>>>(p, y, partial, n);
    const double inv_n = 1.0 / (double)n;
    focal_final_kernel<<<1, NTHREADS, 0, stream>>>(partial, out, nblocks, inv_n);
}